// AllPassWarp_4209067950782
// MI455X (gfx1250) — compile-verified
//
#include <hip/hip_runtime.h>

// CDNA5 / gfx1250, wave32. Matrix engine: V_WMMA_F32_16X16X4_F32.
typedef __attribute__((ext_vector_type(2))) float v2f;
typedef __attribute__((ext_vector_type(8))) float v8f;

struct F3 { float v0, v1, v2; };   // 12-byte triple for LDS b96/2addr ops

#define WMS   30          // warp matrix size
#define NF    90          // features = 3*WMS
#define MAXP  60          // polynomial length (GEMM K)
#define NCOLS 900         // WMS*WMS (GEMM N, flattened (i,j))
#define NT    57          // N tiles: 57*16 = 912 >= 900
#define KT    15          // K tiles: 15*4  = 60
#define WAVES 2           // waves per block
#define SPW   32          // samples per wave = two 16-row M tiles
#define BTILE (KT * 64)   // floats per staged B tile (15 ktiles * 32 lanes * 2)

// ---------------------------------------------------------------------------
// Kernel 1: reorder w_matrix_3d (30,30,60)[i,j,k] into WMMA B-operand layout.
// B tile (ktile,ntile) is 4x16 f32 in 2 VGPRs/lane:
//   lane L<16 : col = ntile*16+L,      v0 -> K=ktile*4+0, v1 -> K=ktile*4+1
//   lane L>=16: col = ntile*16+(L-16), v0 -> K=ktile*4+2, v1 -> K=ktile*4+3
// ---------------------------------------------------------------------------
__global__ void reorder_w_kernel(const float* __restrict__ w3,
                                 float* __restrict__ bw) {
    int idx = blockIdx.x * blockDim.x + threadIdx.x;
    const int total = NT * KT * 32;
    if (idx >= total) return;
    int lane  = idx & 31;
    int tile  = idx >> 5;
    int ktile = tile % KT;
    int ntile = tile / KT;
    int col   = ntile * 16 + (lane & 15);
    int k     = ktile * 4 + ((lane >> 4) << 1);   // +0 or +2
    float v0 = 0.0f, v1 = 0.0f;
    if (col < NCOLS) {
        int i = col / WMS, j = col % WMS;
        const float* p = w3 + ((size_t)(i * WMS + j)) * MAXP + k;
        v0 = p[0];
        v1 = p[1];
    }
    bw[(size_t)idx * 2 + 0] = v0;
    bw[(size_t)idx * 2 + 1] = v1;
}

// ---------------------------------------------------------------------------
// Kernel 2: one wave = 32 samples (two 16-row WMMA M tiles sharing each B).
// B tiles are staged block-wide into double-buffered LDS (read once per block,
// 2x less L2 traffic than per-wave global reads). Warp tiles are consumed in
// VGPRs: warp[m,i,j] scatters x[m,s,i]*w into the per-wave LDS out accumulator.
// ---------------------------------------------------------------------------
__launch_bounds__(WAVES * 32, 1)
__global__ void allpass_warp_kernel(const float* __restrict__ x_in,
                                    const float* __restrict__ alphas,
                                    const float* __restrict__ bw,
                                    float* __restrict__ out,
                                    int n_samples) {
    // s-innermost transposed buffers: index (m*WMS + i)*3 + s
    __shared__ __align__(16) float xT[WAVES][SPW * NF];
    __shared__ __align__(16) float oT[WAVES][SPW * NF];
    __shared__ __align__(16) float Bs[2][BTILE];   // double-buffered B tiles

    const int tid  = threadIdx.x;
    const int lane = tid & 31;
    const int wave = tid >> 5;
    const int sbase = (blockIdx.x * WAVES + wave) * SPW;

    // --- alpha powers -> A operands for both 16-row tiles (60 VGPRs) ---
    // A layout (16x4 f32): lane L<16: M=L, K=kt*4+{0,1}; L>=16: M=L-16, K=kt*4+{2,3}
    v2f A0[KT], A1[KT];
    {
        const float al0 = alphas[sbase + (lane & 15)];
        const float al1 = alphas[sbase + 16 + (lane & 15)];
        float c0 = (lane & 16) ? al0 * al0 : 1.0f;   // a^(2*half)
        float c1 = (lane & 16) ? al1 * al1 : 1.0f;
        const float q0 = (al0 * al0) * (al0 * al0);
        const float q1 = (al1 * al1) * (al1 * al1);
        #pragma unroll
        for (int kt = 0; kt < KT; ++kt) {
            A0[kt][0] = c0; A0[kt][1] = c0 * al0; c0 *= q0;
            A1[kt][0] = c1; A1[kt][1] = c1 * al1; c1 *= q1;
        }
    }

    // --- stage x transposed: lane owns row m=lane (no div/mod) ---
    {
        const float* row = x_in + (size_t)(sbase + lane) * NF;
        #pragma unroll
        for (int s = 0; s < 3; ++s) {
            #pragma unroll
            for (int p = 0; p < KT; ++p) {           // i = 2p, 2p+1
                v2f xv = *(const v2f*)(row + s * WMS + p * 2);
                if (p == 0) xv[0] *= 0.5f;           // leading coeff * 0.5
                int base = (lane * WMS + p * 2) * 3 + s;
                xT[wave][base]     = xv[0];
                xT[wave][base + 3] = xv[1];
            }
        }
        // zero out accumulator: lane owns 90 consecutive floats (45 x b64)
        v2f z = {0.0f, 0.0f};
        float* op = &oT[wave][lane * NF];
        #pragma unroll
        for (int p = 0; p < NF / 2; ++p) *(v2f*)(op + 2 * p) = z;
    }

    // --- stage first B tile block-wide ---
    {
        const float4* src = (const float4*)bw;
        #pragma unroll
        for (int q = 0; q < 4; ++q) {
            int idx = tid + q * 64;
            if (idx < BTILE / 4) *(float4*)&Bs[0][idx * 4] = src[idx];
        }
    }
    __syncthreads();

    const int mhalf = (lane >> 4) << 3;   // C layout: M = r + 8*(L>=16)
    const int csub  = lane & 15;
    int c = csub, i = 0, j = csub;        // c = i*30 + j, incrementally updated

    #pragma unroll 1
    for (int nt = 0; nt < NT; ++nt) {
        const int cur = nt & 1;
        const bool havenext = (nt + 1 < NT);

        // prefetch next B tile into registers (overlaps WMMA below)
        float4 pre0 = {}, pre1 = {}, pre2 = {}, pre3 = {};
        if (havenext) {
            const float4* src = (const float4*)(bw + (size_t)(nt + 1) * BTILE);
            pre0 = src[tid];
            pre1 = src[tid + 64];
            pre2 = src[tid + 128];
            if (tid < BTILE / 4 - 192) pre3 = src[tid + 192];
        }

        // --- two 16x16 warp tiles: 15 x (ds_load_b64 + 2 WMMAs) ---
        v8f acc0 = {}, acc1 = {};
        #pragma unroll
        for (int kt = 0; kt < KT; ++kt) {
            v2f b = *(const v2f*)&Bs[cur][kt * 64 + lane * 2];
            acc0 = __builtin_amdgcn_wmma_f32_16x16x4_f32(
                false, A0[kt], false, b, (short)0, acc0, false, false);
            acc1 = __builtin_amdgcn_wmma_f32_16x16x4_f32(
                false, A1[kt], false, b, (short)0, acc1, false, false);
        }

        // --- apply: warp[m,i,j] -> oT[m,j,*] += xT[m,i,*] * w ---
        if (c < NCOLS) {
            const int xb = (mhalf * WMS + i) * 3;
            const int ob = (mhalf * WMS + j) * 3;
            #pragma unroll
            for (int r = 0; r < 8; ++r) {
                {   // tile 0: samples mhalf + r
                    float w = acc0[r];
                    F3 xv = *(const F3*)&xT[wave][xb + r * (WMS * 3)];
                    F3 ov = *(F3*)&oT[wave][ob + r * (WMS * 3)];
                    ov.v0 += xv.v0 * w; ov.v1 += xv.v1 * w; ov.v2 += xv.v2 * w;
                    *(F3*)&oT[wave][ob + r * (WMS * 3)] = ov;
                }
                {   // tile 1: samples 16 + mhalf + r
                    float w = acc1[r];
                    F3 xv = *(const F3*)&xT[wave][xb + 16 * WMS * 3 + r * (WMS * 3)];
                    F3 ov = *(F3*)&oT[wave][ob + 16 * WMS * 3 + r * (WMS * 3)];
                    ov.v0 += xv.v0 * w; ov.v1 += xv.v1 * w; ov.v2 += xv.v2 * w;
                    *(F3*)&oT[wave][ob + 16 * WMS * 3 + r * (WMS * 3)] = ov;
                }
            }
        }

        __syncthreads();                         // all done reading Bs[cur]
        if (havenext) {
            const int nxt = cur ^ 1;
            *(float4*)&Bs[nxt][tid * 4]         = pre0;
            *(float4*)&Bs[nxt][(tid + 64) * 4]  = pre1;
            *(float4*)&Bs[nxt][(tid + 128) * 4] = pre2;
            if (tid < BTILE / 4 - 192) *(float4*)&Bs[nxt][(tid + 192) * 4] = pre3;
        }
        __syncthreads();                         // Bs[nxt] ready

        c += 16;
        j += 16;
        if (j >= WMS) { j -= WMS; ++i; }         // 16 < 30: at most one carry
    }

    // --- write output un-transposed: lane owns row m=lane (no div/mod) ---
    {
        float* orow = out + (size_t)(sbase + lane) * NF;
        #pragma unroll
        for (int s = 0; s < 3; ++s) {
            #pragma unroll
            for (int p = 0; p < KT; ++p) {
                int base = (lane * WMS + p * 2) * 3 + s;
                v2f ov;
                ov[0] = oT[wave][base];
                ov[1] = oT[wave][base + 3];
                if (p == 0) ov[0] *= 2.0f;       // leading coeff * 2.0
                *(v2f*)(orow + s * WMS + p * 2) = ov;
            }
        }
        // combined_alphas passthrough (second tuple element, appended flat)
        out[(size_t)n_samples * NF + sbase + lane] = alphas[sbase + lane];
    }
}

// ---------------------------------------------------------------------------
extern "C" void kernel_launch(void* const* d_in, const int* in_sizes, int n_in,
                              void* d_out, int out_size, void* d_ws, size_t ws_size,
                              hipStream_t stream) {
    const float* in_tensor = (const float*)d_in[0];   // (T,B,90) f32
    const float* alphas    = (const float*)d_in[1];   // (T,B,1)  f32
    const float* w3        = (const float*)d_in[2];   // (30,30,60) f32
    float* out = (float*)d_out;
    float* bw  = (float*)d_ws;                        // 57*15*32*2 f32 = 219 KB

    const int N = in_sizes[1];                        // T*B samples (128000)

    const int reorder_threads = NT * KT * 32;
    reorder_w_kernel<<<(reorder_threads + 127) / 128, 128, 0, stream>>>(w3, bw);

    const int blocks = N / (WAVES * SPW);             // 2000
    allpass_warp_kernel<<<blocks, WAVES * 32, 0, stream>>>(
        in_tensor, alphas, bw, out, N);
}